// GCNModel_34651796144253
// MI455X (gfx1250) — compile-verified
//
#include <hip/hip_runtime.h>
#include <math.h>

typedef __attribute__((ext_vector_type(2))) float v2f;
typedef __attribute__((ext_vector_type(8))) float v8f;

#define KDIM 128

// ---------------------------------------------------------------------------
// C[rows x M] = A[rows x 128] * W[128 x M]; rows % 16 == 0, M % 64 == 0.
// One 128-thread block (4 waves, wave32) -> one 16x64 C tile.
// Each wave: 16x16 tile, 32x V_WMMA_F32_16X16X4_F32 over K=128.
// ---------------------------------------------------------------------------
__global__ __launch_bounds__(128) void gemm_wmma_k128(
    const float* __restrict__ A, const float* __restrict__ W,
    float* __restrict__ C, int M)
{
    __shared__ float As[16][KDIM + 4];   // +4 pad: 16B-aligned rows, bank spread
    __shared__ float Ws[KDIM][64 + 4];

    const int tid  = threadIdx.x;
    const int wave = tid >> 5;
    const int lane = tid & 31;
    const int row0 = blockIdx.x << 4;    // 16-row tile
    const int col0 = blockIdx.y << 6;    // 64-col tile

    // Stage A tile (16x128 f32) with float4 loads
#pragma unroll
    for (int i = 0; i < 4; ++i) {
        int l4 = tid + i * 128;
        int r  = l4 >> 5;
        int k4 = l4 & 31;
        float4 v = *reinterpret_cast<const float4*>(
            A + (size_t)(row0 + r) * KDIM + k4 * 4);
        *reinterpret_cast<float4*>(&As[r][k4 * 4]) = v;
    }
    // Stage W tile (128x64 f32) with float4 loads
#pragma unroll
    for (int i = 0; i < 16; ++i) {
        int l4 = tid + i * 128;
        int k  = l4 >> 4;
        int n4 = l4 & 15;
        float4 v = *reinterpret_cast<const float4*>(
            W + (size_t)k * M + col0 + n4 * 4);
        *reinterpret_cast<float4*>(&Ws[k][n4 * 4]) = v;
    }
    __syncthreads();

    // WMMA lane layout (ISA 7.12.2):
    //   A: lane holds {A[m][k0], A[m][k0+1]}, m = lane%16, k0 = 2*(lane/16)
    //   B: lane holds {B[k0][n],  B[k0+1][n]}, n = lane%16
    const int mrow = lane & 15;
    const int k0   = (lane >> 4) << 1;
    const int nl   = (wave << 4) + (lane & 15);

    v8f acc = {0.f, 0.f, 0.f, 0.f, 0.f, 0.f, 0.f, 0.f};
#pragma unroll
    for (int kk = 0; kk < KDIM; kk += 4) {
        v2f a, b;
        a.x = As[mrow][kk + k0];
        a.y = As[mrow][kk + k0 + 1];
        b.x = Ws[kk + k0][nl];
        b.y = Ws[kk + k0 + 1][nl];
        acc = __builtin_amdgcn_wmma_f32_16x16x4_f32(
            false, a, false, b, (short)0, acc, false, false);
    }

    // C/D layout: VGPR r -> M=r (lanes 0-15) / M=r+8 (lanes 16-31)
    const int rbase = row0 + ((lane >> 4) << 3);
#pragma unroll
    for (int r = 0; r < 8; ++r)
        C[(size_t)(rbase + r) * M + col0 + nl] = acc[r];
}

// ---------------------------------------------------------------------------
// Degree / normalization helpers
// ---------------------------------------------------------------------------
__global__ void fill_f32(float* __restrict__ p, float v, int n)
{
    int i = blockIdx.x * blockDim.x + threadIdx.x;
    if (i < n) p[i] = v;
}

__global__ void deg_accum(const int* __restrict__ dst, float* __restrict__ deg, int E)
{
    int e = blockIdx.x * blockDim.x + threadIdx.x;
    if (e < E) atomicAdd(&deg[dst[e]], 1.0f);
}

__global__ void to_rsqrt(float* __restrict__ d, int n)
{
    int i = blockIdx.x * blockDim.x + threadIdx.x;
    if (i < n) d[i] = rsqrtf(d[i]);
}

__global__ void init_scalars(float* __restrict__ scal)
{
    scal[0] = -__builtin_inff();  // running max
    scal[1] = 0.0f;               // running sum
}

// out[i][:] = bias + t[i][:] * dis[i]^2   (self-loop message + bias)
__global__ void self_init(const float4* __restrict__ t, const float* __restrict__ dis,
                          const float* __restrict__ bias, float4* __restrict__ out,
                          int total4, int qshift)
{
    int idx = blockIdx.x * blockDim.x + threadIdx.x;
    if (idx >= total4) return;
    int i = idx >> qshift;
    int j = idx & ((1 << qshift) - 1);
    float d  = dis[i];
    float dd = d * d;
    float4 tv = t[idx];
    float4 bv = reinterpret_cast<const float4*>(bias)[j];
    out[idx] = make_float4(bv.x + tv.x * dd, bv.y + tv.y * dd,
                           bv.z + tv.z * dd, bv.w + tv.w * dd);
}

// per edge: out[dst][:] += t[src][:] * dis[src]*dis[dst]
__global__ void edge_scatter(const int* __restrict__ src, const int* __restrict__ dst,
                             const float4* __restrict__ t, const float* __restrict__ dis,
                             float* __restrict__ out, int E, int M, int qshift)
{
    int gid = blockIdx.x * blockDim.x + threadIdx.x;
    int total = E << qshift;
    if (gid >= total) return;
    int e = gid >> qshift;
    int c = gid & ((1 << qshift) - 1);
    int s = src[e];
    int d = dst[e];
    float nrm = dis[s] * dis[d];
    float4 tv = t[((size_t)s << qshift) + c];
    float* o = out + (size_t)d * M + c * 4;
    atomicAdd(o + 0, tv.x * nrm);
    atomicAdd(o + 1, tv.y * nrm);
    atomicAdd(o + 2, tv.z * nrm);
    atomicAdd(o + 3, tv.w * nrm);
}

__global__ void relu4(float4* __restrict__ x, int n4)
{
    int i = blockIdx.x * blockDim.x + threadIdx.x;
    if (i >= n4) return;
    float4 v = x[i];
    x[i] = make_float4(fmaxf(v.x, 0.f), fmaxf(v.y, 0.f),
                       fmaxf(v.z, 0.f), fmaxf(v.w, 0.f));
}

// ---------------------------------------------------------------------------
// Attention head (H3 = 64): wave-per-node dot products + global softmax
// ---------------------------------------------------------------------------
__device__ inline void atomicMaxFloat(float* addr, float val)
{
    if (val >= 0.0f) atomicMax((int*)addr, __float_as_int(val));
    else             atomicMin((unsigned int*)addr, __float_as_uint(val));
}

__global__ __launch_bounds__(128) void attn_score(
    const float* __restrict__ h, const float* __restrict__ aw,
    const float* __restrict__ ab, float* __restrict__ s,
    float* __restrict__ gmax, int n)
{
    const int wave = threadIdx.x >> 5;
    const int lane = threadIdx.x & 31;
    const int node = blockIdx.x * 4 + wave;
    float acc = 0.f;
    if (node < n) {
        const float* hr = h + (size_t)node * 64;
        acc = hr[lane] * aw[lane] + hr[32 + lane] * aw[32 + lane];
    }
#pragma unroll
    for (int o = 16; o > 0; o >>= 1) acc += __shfl_xor(acc, o, 32);
    __shared__ float red[4];
    if (lane == 0) {
        float sc = acc + ab[0];
        if (node < n) s[node] = sc;
        red[wave] = (node < n) ? sc : -__builtin_inff();
    }
    __syncthreads();
    if (threadIdx.x == 0) {
        float m = fmaxf(fmaxf(red[0], red[1]), fmaxf(red[2], red[3]));
        atomicMaxFloat(gmax, m);
    }
}

__global__ __launch_bounds__(256) void exp_sum(
    const float* __restrict__ s, const float* __restrict__ gmax,
    float* __restrict__ e, float* __restrict__ gsum, int n)
{
    int i = blockIdx.x * blockDim.x + threadIdx.x;
    float m = *gmax;
    float v = 0.f;
    if (i < n) {
        v = __expf(s[i] - m);
        e[i] = v;
    }
#pragma unroll
    for (int o = 16; o > 0; o >>= 1) v += __shfl_xor(v, o, 32);
    __shared__ float red[8];
    int w = threadIdx.x >> 5, l = threadIdx.x & 31;
    if (l == 0) red[w] = v;
    __syncthreads();
    if (threadIdx.x == 0) {
        float t = 0.f;
#pragma unroll
        for (int k = 0; k < 8; ++k) t += red[k];
        atomicAdd(gsum, t);
    }
}

__global__ __launch_bounds__(128) void finalize(
    const float* __restrict__ h, const float* __restrict__ e,
    const float* __restrict__ gsum, const float* __restrict__ fw,
    const float* __restrict__ fb, float* __restrict__ out,
    float* __restrict__ attn_out, int n)
{
    const int wave = threadIdx.x >> 5;
    const int lane = threadIdx.x & 31;
    const int node = blockIdx.x * 4 + wave;
    if (node >= n) return;
    const float* hr = h + (size_t)node * 64;
    float v = hr[lane] * fw[lane] + hr[32 + lane] * fw[32 + lane];
#pragma unroll
    for (int o = 16; o > 0; o >>= 1) v += __shfl_xor(v, o, 32);
    if (lane == 0) {
        float a = e[node] / gsum[0];
        float z = a * v + fb[0];
        out[node]      = 1.f / (1.f + __expf(-z));
        attn_out[node] = a;
    }
}

// ---------------------------------------------------------------------------
extern "C" void kernel_launch(void* const* d_in, const int* in_sizes, int n_in,
                              void* d_out, int out_size, void* d_ws, size_t ws_size,
                              hipStream_t stream)
{
    (void)n_in; (void)out_size; (void)ws_size;
    const float* x  = (const float*)d_in[0];
    const int*   ei = (const int*)  d_in[1];
    const float* W1 = (const float*)d_in[2];
    const float* b1 = (const float*)d_in[3];
    const float* W2 = (const float*)d_in[4];
    const float* b2 = (const float*)d_in[5];
    const float* W3 = (const float*)d_in[6];
    const float* b3 = (const float*)d_in[7];
    const float* aW = (const float*)d_in[8];
    const float* ab = (const float*)d_in[9];
    const float* fW = (const float*)d_in[10];
    const float* fb = (const float*)d_in[11];

    const int N = in_sizes[0] / 128;
    const int E = in_sizes[1] / 2;
    const int* src = ei;
    const int* dst = ei + E;

    char* p = (char*)d_ws;
    auto carve = [&](size_t bytes) {
        char* r = p;
        p += (bytes + 255) & ~(size_t)255;
        return r;
    };
    float* dis  = (float*)carve((size_t)N * 4);
    float* T    = (float*)carve((size_t)N * 128 * 4);
    float* AGG  = (float*)carve((size_t)N * 128 * 4);
    float* H    = (float*)carve((size_t)N * 128 * 4);
    float* sbuf = (float*)carve((size_t)N * 4);
    float* ebuf = (float*)carve((size_t)N * 4);
    float* scal = (float*)carve(256);

    float* outp  = (float*)d_out;
    float* attnp = outp + N;

    const int TB = 256;
    auto blk = [](int n, int t) { return (n + t - 1) / t; };

    // deg -> dis = rsqrt(deg); self-loops contribute the init 1.0
    fill_f32   <<<blk(N, TB), TB, 0, stream>>>(dis, 1.0f, N);
    deg_accum  <<<blk(E, TB), TB, 0, stream>>>(dst, dis, E);
    to_rsqrt   <<<blk(N, TB), TB, 0, stream>>>(dis, N);
    init_scalars<<<1, 1, 0, stream>>>(scal);

    // ---- layer 1: x(128) -> AGG(128)
    gemm_wmma_k128<<<dim3(N / 16, 2), 128, 0, stream>>>(x, W1, T, 128);
    self_init   <<<blk(N * 32, TB), TB, 0, stream>>>((const float4*)T, dis, b1, (float4*)AGG, N * 32, 5);
    edge_scatter<<<blk(E << 5, TB), TB, 0, stream>>>(src, dst, (const float4*)T, dis, AGG, E, 128, 5);
    relu4       <<<blk(N * 32, TB), TB, 0, stream>>>((float4*)AGG, N * 32);

    // ---- layer 2: AGG(128) -> H(128)
    gemm_wmma_k128<<<dim3(N / 16, 2), 128, 0, stream>>>(AGG, W2, T, 128);
    self_init   <<<blk(N * 32, TB), TB, 0, stream>>>((const float4*)T, dis, b2, (float4*)H, N * 32, 5);
    edge_scatter<<<blk(E << 5, TB), TB, 0, stream>>>(src, dst, (const float4*)T, dis, H, E, 128, 5);
    relu4       <<<blk(N * 32, TB), TB, 0, stream>>>((float4*)H, N * 32);

    // ---- layer 3: H(128) -> AGG(64)
    gemm_wmma_k128<<<dim3(N / 16, 1), 128, 0, stream>>>(H, W3, T, 64);
    self_init   <<<blk(N * 16, TB), TB, 0, stream>>>((const float4*)T, dis, b3, (float4*)AGG, N * 16, 4);
    edge_scatter<<<blk(E << 4, TB), TB, 0, stream>>>(src, dst, (const float4*)T, dis, AGG, E, 64, 4);
    relu4       <<<blk(N * 16, TB), TB, 0, stream>>>((float4*)AGG, N * 16);

    // ---- attention softmax over nodes + sigmoid head
    attn_score<<<blk(N, 4), 128, 0, stream>>>(AGG, aW, ab, sbuf, &scal[0], N);
    exp_sum   <<<blk(N, 256), 256, 0, stream>>>(sbuf, &scal[0], ebuf, &scal[1], N);
    finalize  <<<blk(N, 4), 128, 0, stream>>>(AGG, ebuf, &scal[1], fW, fb, outp, attnp, N);
}